// PyramidNSMLayer_77627238908017
// MI455X (gfx1250) — compile-verified
//
#include <hip/hip_runtime.h>
#include <stdint.h>

typedef __attribute__((ext_vector_type(2))) float v2f;
typedef __attribute__((ext_vector_type(8))) float v8f;

#define N_TOTAL 3133440
#define OFF1 2359296
#define OFF2 2949120
#define OFF3 3096576
#define TOPK 4000
#define CPAD 4096
#define NMSB 512
#define NUM_ROIS 300

// monotone float->uint key (scores are >=0, but be general)
__device__ __forceinline__ uint32_t score_key(float s) {
  uint32_t u = __float_as_uint(s);
  uint32_t m = (uint32_t)(((int32_t)u) >> 31) | 0x80000000u;
  return u ^ m;
}

__device__ __forceinline__ float load_score(int i, const float* x1, const float* x2,
                                            const float* x3, const float* x4) {
  if (i < OFF1) return x1[(size_t)i * 6];
  if (i < OFF2) return x2[(size_t)(i - OFF1) * 6];
  if (i < OFF3) return x3[(size_t)(i - OFF2) * 6];
  return x4[(size_t)(i - OFF3) * 6];
}

// ---------------- init ----------------
__global__ void init_kernel(uint32_t* hist1, uint32_t* hist2, int* meta, uint32_t* cnt,
                            float* cscore, int* cidx, float* cbox) {
  int i = blockIdx.x * blockDim.x + threadIdx.x;
  if (i < 4096) {
    hist1[i] = 0u;
    hist2[i] = 0u;
    cscore[i] = -1.0f;
    cidx[i] = 0x7FFFFFFF;
  }
  if (i < 16384) cbox[i] = 0.0f;
  if (i < 16) meta[i] = 0;
  if (i < 4) cnt[i] = 0u;
}

// ---------------- level-1 histogram (bits 31:20 of key), LDS privatized ----------------
__global__ __launch_bounds__(512) void hist_kernel(const float* __restrict__ x1,
                                                   const float* __restrict__ x2,
                                                   const float* __restrict__ x3,
                                                   const float* __restrict__ x4,
                                                   uint32_t* __restrict__ hist1) {
  __shared__ uint32_t lh[4096];
  for (int b = threadIdx.x; b < 4096; b += 512) lh[b] = 0u;
  __syncthreads();
  int i = blockIdx.x * 512 + threadIdx.x;
  if (i < N_TOTAL) {
    float s = load_score(i, x1, x2, x3, x4);
    uint32_t key = score_key(s);
    atomicAdd(&lh[key >> 20], 1u);
  }
  __syncthreads();
  for (int b = threadIdx.x; b < 4096; b += 512) {
    uint32_t v = lh[b];
    if (v) atomicAdd(&hist1[b], v);
  }
}

// ---------------- descending suffix-scan over 4096 bins, find crossing bin ----------------
// level 0: target = TOPK, writes meta[0..2] = {bin, countAbove, need}
// level 1: target = meta[2], writes meta[3..5]
__global__ void scan_kernel(const uint32_t* __restrict__ hist, int* meta, int level) {
  __shared__ uint32_t pre[256];
  int t = threadIdx.x;
  uint32_t target = (level == 0) ? (uint32_t)TOPK : (uint32_t)meta[2];
  uint32_t loc[16];
  uint32_t s = 0;
  for (int k = 0; k < 16; k++) {
    loc[k] = hist[4095 - (t * 16 + k)];
    s += loc[k];
  }
  pre[t] = s;
  __syncthreads();
  for (int off = 1; off < 256; off <<= 1) {
    uint32_t add = (t >= off) ? pre[t - off] : 0u;
    __syncthreads();
    pre[t] += add;
    __syncthreads();
  }
  uint32_t excl = pre[t] - s;
  int base = level * 3;
  uint32_t cum = excl;
  for (int k = 0; k < 16; k++) {
    uint32_t c = loc[k];
    if (cum < target && cum + c >= target) {
      meta[base + 0] = 4095 - (t * 16 + k);
      meta[base + 1] = (int)cum;              // strictly-above count
      meta[base + 2] = (int)(target - cum);   // need from this bin
    }
    cum += c;
  }
  if (t == 255 && cum < target) {  // total < target: take everything
    meta[base + 0] = 0;
    meta[base + 1] = (int)(cum - loc[15]);
    meta[base + 2] = (int)loc[15];
  }
}

// ---------------- level-2 histogram (bits 19:8) for elements in level-1 bin ----------------
__global__ void hist2_kernel(const float* __restrict__ x1, const float* __restrict__ x2,
                             const float* __restrict__ x3, const float* __restrict__ x4,
                             const int* __restrict__ meta, uint32_t* __restrict__ hist2) {
  int i = blockIdx.x * blockDim.x + threadIdx.x;
  if (i >= N_TOTAL) return;
  uint32_t B1 = (uint32_t)meta[0];
  float s = load_score(i, x1, x2, x3, x4);
  uint32_t key = score_key(s);
  if ((key >> 20) == B1) atomicAdd(&hist2[(key >> 8) & 0xFFFu], 1u);
}

// ---------------- compact top-K survivors, full box decode only for survivors ----------------
__global__ void compact_kernel(const float* x1, const float* x2, const float* x3, const float* x4,
                               const float* an1, const float* an2, const float* an3, const float* an4,
                               const int* __restrict__ meta, uint32_t* cnt,
                               float* __restrict__ cbox, float* __restrict__ cscore,
                               int* __restrict__ cidxArr) {
  int i = blockIdx.x * blockDim.x + threadIdx.x;
  if (i >= N_TOTAL) return;
  const float *xp, *ap;
  int a;
  if (i < OFF1)      { xp = x1; ap = an1; a = i; }
  else if (i < OFF2) { xp = x2; ap = an2; a = i - OFF1; }
  else if (i < OFF3) { xp = x3; ap = an3; a = i - OFF2; }
  else               { xp = x4; ap = an4; a = i - OFF3; }

  float sc = xp[(size_t)a * 6];
  uint32_t key = score_key(sc);
  uint32_t cut = (((uint32_t)meta[0]) << 12) | ((uint32_t)meta[3]);
  uint32_t k24 = key >> 8;
  int slot = -1;
  if (k24 > cut) {
    slot = (int)atomicAdd(&cnt[0], 1u);
  } else if (k24 == cut) {
    uint32_t tix = atomicAdd(&cnt[1], 1u);
    if (tix < (uint32_t)meta[5]) slot = meta[1] + meta[4] + (int)tix;
  }
  if (slot < 0 || slot >= TOPK) return;

  float dy = xp[(size_t)a * 6 + 2], dxv = xp[(size_t)a * 6 + 3];
  float dh = xp[(size_t)a * 6 + 4], dw  = xp[(size_t)a * 6 + 5];
  float ay1 = ap[(size_t)a * 4 + 0], ax1 = ap[(size_t)a * 4 + 1];
  float ay2 = ap[(size_t)a * 4 + 2], ax2 = ap[(size_t)a * 4 + 3];
  float ah = ay2 - ay1, aw = ax2 - ax1;
  float acy = ay1 + 0.5f * ah, acx = ax1 + 0.5f * aw;
  float cy = acy + dy * ah, cx = acx + dxv * aw;
  float bh = ah * expf(dh), bw = aw * expf(dw);
  float y1 = fminf(fmaxf(cy - 0.5f * bh, 0.0f), 2048.0f);
  float xx = fminf(fmaxf(cx - 0.5f * bw, 0.0f), 2048.0f);
  float y2 = fminf(fmaxf(cy + 0.5f * bh, 0.0f), 2048.0f);
  float x2c = fminf(fmaxf(cx + 0.5f * bw, 0.0f), 2048.0f);
  cbox[(size_t)slot * 4 + 0] = y1;
  cbox[(size_t)slot * 4 + 1] = xx;
  cbox[(size_t)slot * 4 + 2] = y2;
  cbox[(size_t)slot * 4 + 3] = x2c;
  cscore[slot] = sc;
  cidxArr[slot] = i;
}

// ---------------- single-workgroup greedy NMS, candidates LDS-resident ----------------
__global__ __launch_bounds__(NMSB) void nms_kernel(const float* __restrict__ cbox,
                                                   const float* __restrict__ cscore,
                                                   const int* __restrict__ cidx,
                                                   float* __restrict__ out) {
  __shared__ float sB[CPAD * 4];
  __shared__ float sS[CPAD];
  __shared__ float sA[CPAD];
  __shared__ int   sI[CPAD];
  __shared__ float rS[NMSB];
  __shared__ int   rI[NMSB];
  __shared__ float sBest[5];  // y1,x1,y2,x2,area
  int tid = threadIdx.x;

  // Stage candidate boxes global -> LDS via async-to-LDS (ASYNCcnt path).
  for (int c = tid; c < CPAD; c += NMSB) {
    uint32_t ldst = (uint32_t)(uintptr_t)(&sB[c * 4]);
    uint64_t ga = (uint64_t)(uintptr_t)(cbox + (size_t)c * 4);
    asm volatile("global_load_async_to_lds_b128 %0, %1, off" ::"v"(ldst), "v"(ga) : "memory");
    float sc = cscore[c];
    sS[c] = (sc >= 0.5f) ? sc : -1.0f;  // SCORE_THR
    sI[c] = cidx[c];
  }
  asm volatile("s_wait_asynccnt 0x0" ::: "memory");
  __syncthreads();

  // Areas via WMMA: D = A(16x4 selector) * B(4x16 box coords).
  // A row0 = (-1,0,1,0) -> h ; A row1 = (0,-1,0,1) -> w ; D lands h,w of box n in lane n.
  int lane = tid & 31;
  int wave = tid >> 5;
  for (int t = wave; t < CPAD / 16; t += NMSB / 32) {
    int n = lane & 15;
    bool lo = lane < 16;  // lanes 0-15 carry K=0,1 ; lanes 16-31 carry K=2,3
    v2f A;
    A[0] = (n == 0) ? (lo ? -1.0f : 1.0f) : 0.0f;
    A[1] = (n == 1) ? (lo ? -1.0f : 1.0f) : 0.0f;
    int cp = lo ? 0 : 2;
    int bx = (t * 16 + n) * 4;
    v2f Bm;
    Bm[0] = sB[bx + cp];
    Bm[1] = sB[bx + cp + 1];
    v8f C = {};
#if __has_builtin(__builtin_amdgcn_wmma_f32_16x16x4_f32)
    v8f D = __builtin_amdgcn_wmma_f32_16x16x4_f32(false, A, false, Bm, (short)0, C, false, false);
    if (lo) sA[t * 16 + n] = D[0] * D[1];
#else
    (void)A; (void)Bm; (void)C;
    if (lo) {
      float h = sB[bx + 2] - sB[bx + 0];
      float w = sB[bx + 3] - sB[bx + 1];
      sA[t * 16 + n] = h * w;
    }
#endif
  }
  __syncthreads();

  for (int r = 0; r < NUM_ROIS; r++) {
    // parallel argmax, tie-break on smallest original index (matches top_k order + argmax)
    float ms = -2.0f;
    int mslot = 0;
    for (int c = tid; c < CPAD; c += NMSB) {
      float s = sS[c];
      if (s > ms || (s == ms && sI[c] < sI[mslot])) { ms = s; mslot = c; }
    }
    rS[tid] = ms;
    rI[tid] = mslot;
    __syncthreads();
    for (int st = NMSB / 2; st > 0; st >>= 1) {
      if (tid < st) {
        float s1 = rS[tid], s2 = rS[tid + st];
        int c1 = rI[tid], c2 = rI[tid + st];
        if (s2 > s1 || (s2 == s1 && sI[c2] < sI[c1])) { rS[tid] = s2; rI[tid] = c2; }
      }
      __syncthreads();
    }
    if (tid == 0) {
      int b = rI[0];
      float bs = rS[0];
      float y1 = sB[b * 4 + 0], x1 = sB[b * 4 + 1];
      float y2 = sB[b * 4 + 2], x2 = sB[b * 4 + 3];
      sBest[0] = y1; sBest[1] = x1; sBest[2] = y2; sBest[3] = x2; sBest[4] = sA[b];
      bool valid = bs > 0.0f;
      out[r * 4 + 0] = valid ? y1 : 0.0f;
      out[r * 4 + 1] = valid ? x1 : 0.0f;
      out[r * 4 + 2] = valid ? y2 : 0.0f;
      out[r * 4 + 3] = valid ? x2 : 0.0f;
      sS[b] = -1.0f;
    }
    __syncthreads();
    float by1 = sBest[0], bx1 = sBest[1], by2 = sBest[2], bx2 = sBest[3], ba = sBest[4];
    for (int c = tid; c < CPAD; c += NMSB) {
      float yy1 = fmaxf(by1, sB[c * 4 + 0]);
      float xx1 = fmaxf(bx1, sB[c * 4 + 1]);
      float yy2 = fminf(by2, sB[c * 4 + 2]);
      float xx2 = fminf(bx2, sB[c * 4 + 3]);
      float inter = fmaxf(yy2 - yy1, 0.0f) * fmaxf(xx2 - xx1, 0.0f);
      float iou = inter / (ba + sA[c] - inter + 1e-9f);
      if (iou > 0.5f) sS[c] = -1.0f;
    }
    __syncthreads();
  }
}

extern "C" void kernel_launch(void* const* d_in, const int* in_sizes, int n_in,
                              void* d_out, int out_size, void* d_ws, size_t ws_size,
                              hipStream_t stream) {
  (void)in_sizes; (void)n_in; (void)out_size;
  const float* x1 = (const float*)d_in[0];
  const float* a1 = (const float*)d_in[1];
  const float* x2 = (const float*)d_in[2];
  const float* a2 = (const float*)d_in[3];
  const float* x3 = (const float*)d_in[4];
  const float* a3 = (const float*)d_in[5];
  const float* x4 = (const float*)d_in[6];
  const float* a4 = (const float*)d_in[7];
  float* out = (float*)d_out;

  // workspace layout (32-bit words)
  uint32_t* w = (uint32_t*)d_ws;
  uint32_t* hist1 = w;                    // 4096
  uint32_t* hist2 = w + 4096;             // 4096
  int* meta = (int*)(w + 8192);           // 16
  uint32_t* cnt = w + 8208;               // 16
  float* cscore = (float*)(w + 8224);     // 4096
  int* cidx = (int*)(w + 8224 + 4096);    // 4096
  float* cbox = (float*)(w + 8224 + 8192);// 16384
  if (ws_size < (size_t)(8224 + 8192 + 16384) * 4) return;

  init_kernel<<<64, 256, 0, stream>>>(hist1, hist2, meta, cnt, cscore, cidx, cbox);
  hist_kernel<<<(N_TOTAL + 511) / 512, 512, 0, stream>>>(x1, x2, x3, x4, hist1);
  scan_kernel<<<1, 256, 0, stream>>>(hist1, meta, 0);
  hist2_kernel<<<(N_TOTAL + 255) / 256, 256, 0, stream>>>(x1, x2, x3, x4, meta, hist2);
  scan_kernel<<<1, 256, 0, stream>>>(hist2, meta, 1);
  compact_kernel<<<(N_TOTAL + 255) / 256, 256, 0, stream>>>(x1, x2, x3, x4, a1, a2, a3, a4,
                                                            meta, cnt, cbox, cscore, cidx);
  nms_kernel<<<1, NMSB, 0, stream>>>(cbox, cscore, cidx, out);
}